// MonarchMatrix_5111011082510
// MI455X (gfx1250) — compile-verified
//
#include <hip/hip_runtime.h>

// ---- CDNA5 wave32 vector types ----
typedef float v2f __attribute__((ext_vector_type(2)));
typedef float v8f __attribute__((ext_vector_type(8)));
typedef unsigned int v4u __attribute__((ext_vector_type(4)));
typedef int v4i __attribute__((ext_vector_type(4)));
typedef int v8i __attribute__((ext_vector_type(8)));

#define S        128          // block size (sqrt of 16384)
#define NN       16384        // S*S, per-row vector length
#define TB       16           // batch rows per workgroup
#define BG       16           // blocks per workgroup
#define XS_STRIDE 272         // k-row stride in dwords: 16*16 + TDM pad (1 dw / 16 dw)
#define OS_BSTR   2067        // padded per-b stride in dwords (16*129 + 3)
#define OS_BLSTR  129         // padded per-bl stride in dwords
#define XS_DWORDS (S * XS_STRIDE + 32)   // + margin past last TDM pad
#define OS_DWORDS (TB * OS_BSTR)

// ---------------------------------------------------------------------------
// TDM: load a 3D tile  [dim0 = 16 contiguous elems (bl)] x
//                      [dim1 = 16 lines @ stride 16384  (b)] x
//                      [dim2 = 128 lines @ stride 128   (k)]
// into LDS with 1 dword of padding per 16 dwords, producing the layout
//   xs[k*272 + b*17 + bl]   (bank-conflict-free for the WMMA B-fragment reads)
// Note ISA stride naming: dim1 uses tensor_dim0_stride, dim2 uses
// tensor_dim1_stride (off-by-one vs the dim index).
// ---------------------------------------------------------------------------
__device__ __forceinline__ void tdm_load_tile(const float* gbase, unsigned lds_off)
{
    const unsigned long long ga = (unsigned long long)(uintptr_t)gbase;

    v4u g0;
    g0.x = 1u;                                   // count=1, user descriptor
    g0.y = lds_off;                              // lds_addr (bytes, wave-relative)
    g0.z = (unsigned)(ga & 0xFFFFFFFFu);         // global_addr[31:0]
    g0.w = (unsigned)((ga >> 32) & 0x1FFFFFFu)   // global_addr[56:32]
         | (2u << 30);                           // type = 2 ("image")

    v8i g1;
    g1[0] = (int)((2u << 16)                     // data_size = 4 bytes
                | (1u << 20)                     // pad_enable
                | (3u << 22)                     // pad_interval: 16 dwords
                | (0u << 25));                   // pad_amount: 1 dword
    g1[1] = (int)(16u << 16);                    // tensor_dim0 = 16  (bl extent)
    g1[2] = (int)(16u << 16);                    // dim0 hi=0 | tensor_dim1 = 16 (b)
    g1[3] = (int)(16u << 16);                    // dim1 hi=0 | tile_dim0 = 16
    g1[4] = (int)(16u | (128u << 16));           // tile_dim1 = 16, tile_dim2 = 128
    g1[5] = (int)NN;                             // tensor_dim0_stride = 16384 (b)
    g1[6] = (int)(128u << 16);                   // str0 hi=0 | tensor_dim1_stride = 128 (k)
    g1[7] = 0;                                   // str1 hi

    v4i g2;
    g2[0] = 128;                                 // tensor_dim2 = 128 (k extent)
    g2[1] = 1;                                   // tensor_dim3 (unused)
    g2[2] = 0;                                   // tensor_dim2_stride (dim3 unused)
    g2[3] = 0;                                   // str2 hi | tile_dim3 = 0

    v4i g3 = {0, 0, 0, 0};                       // dim4 unused

    v8i g4 = {0, 0, 0, 0, 0, 0, 0, 0};           // extra group (clang-23 6-arg form)

    __builtin_amdgcn_tensor_load_to_lds(g0, g1, g2, g3, g4, 0);
}

// ---------------------------------------------------------------------------
// Generic Monarch stage (input gather is always "transposed": blk fastest):
//   O_idx(b,blk,n) = TRANS_OUT ? b*NN + n*S + blk : b*NN + blk*S + n
//   O[...] = sum_k M[blk, n, k] * I[b, k*S + blk]
// Stage 1: (x, L, TRANS_OUT=0) -> T[b][m][j]
// Stage 2: (T, R, TRANS_OUT=1) -> out[b][i*128+j]
// ---------------------------------------------------------------------------
template <bool TRANS_OUT, bool NT_OUT>
__global__ __launch_bounds__(256)
void monarch_stage(const float* __restrict__ in, const float* __restrict__ mat,
                   float* __restrict__ out)
{
    __shared__ float xs[XS_DWORDS];  // input tile via TDM: xs[k*272 + b*17 + bl]
    __shared__ float os[OS_DWORDS];  // output tile: os[b*2067 + bl*129 + n]

    const int tid  = threadIdx.x;
    const long b0  = (long)blockIdx.x * TB;
    const int blk0 = blockIdx.y * BG;

    // ---------- Phase 1: TDM async tile load, global -> LDS ----------
    if (tid < 32) {                              // wave 0 issues the DMA
        tdm_load_tile(in + b0 * NN + blk0, (unsigned)(uintptr_t)xs);
        __builtin_amdgcn_s_wait_tensorcnt(0);
    }
    __syncthreads();

    // ---------- Phase 2: WMMA compute ----------
    const int wave = tid >> 5;
    const int lane = tid & 31;
    const int half = lane >> 4;   // half-wave selects K pair / M-half
    const int lr   = lane & 15;

    for (int bi = 0; bi < 2; ++bi) {
        const int bl  = wave * 2 + bi;           // local block
        const int blk = blk0 + bl;               // global block index

        // A-frag (mat): lane holds row M = lr (+16 per n-tile), K pair 2*half
        const float* matBase = mat + (long)blk * (S * S) + lr * S + 2 * half;
        // B-frag (tile): VGPR0 = xs[k4+2*half][lr][bl], VGPR1 = +1 in k
        const float* xsBase = xs + (2 * half) * XS_STRIDE + lr * 17 + bl;

        v8f acc[8];
        #pragma unroll
        for (int t = 0; t < 8; ++t)
            #pragma unroll
            for (int v = 0; v < 8; ++v) acc[t][v] = 0.0f;

        #pragma unroll 2
        for (int kk = 0; kk < 32; ++kk) {
            const int k4 = kk * 4;
            v2f bfrag;
            bfrag.x = xsBase[(k4 + 0) * XS_STRIDE];
            bfrag.y = xsBase[(k4 + 1) * XS_STRIDE];
            #pragma unroll
            for (int t = 0; t < 8; ++t) {
                const v2f afrag = *(const v2f*)(matBase + t * 16 * S + k4);
                acc[t] = __builtin_amdgcn_wmma_f32_16x16x4_f32(
                    false, afrag, false, bfrag,
                    (short)0, acc[t], false, false);
            }
        }

        // D tile: lane lr = col b; VGPR v = row n = t*16 + v + 8*half
        float* osp = os + lr * OS_BSTR + bl * OS_BLSTR + 8 * half;
        #pragma unroll
        for (int t = 0; t < 8; ++t)
            #pragma unroll
            for (int v = 0; v < 8; ++v)
                osp[t * 16 + v] = acc[t][v];
    }
    __syncthreads();

    // ---------- Phase 3: cooperative LDS -> global ----------
    if (!TRANS_OUT) {
        // O[b, (blk0+bl)*S + n]: lanes span n -> fully contiguous 512B runs
        float* obase = out + b0 * NN + (long)blk0 * S;
        #pragma unroll 4
        for (int idx = tid; idx < TB * S * BG; idx += 256) {
            const int n  = idx & 127;
            const int t  = idx >> 7;
            const int bl = t & 15;
            const int b  = t >> 4;
            float* p = obase + (long)b * NN + bl * S + n;
            const float val = os[b * OS_BSTR + bl * OS_BLSTR + n];
            if (NT_OUT) __builtin_nontemporal_store(val, p);
            else        *p = val;
        }
    } else {
        // O[b, n*S + blk0+bl]: lanes span bl -> 64B segments
        float* obase = out + b0 * NN + blk0;
        #pragma unroll 4
        for (int idx = tid; idx < TB * S * BG; idx += 256) {
            const int bl = idx & 15;
            const int t  = idx >> 4;
            const int b  = t & 15;
            const int n  = t >> 4;
            float* p = obase + (long)b * NN + n * S + bl;
            const float val = os[b * OS_BSTR + bl * OS_BLSTR + n];
            if (NT_OUT) __builtin_nontemporal_store(val, p);
            else        *p = val;
        }
    }
}

extern "C" void kernel_launch(void* const* d_in, const int* in_sizes, int n_in,
                              void* d_out, int out_size, void* d_ws, size_t ws_size,
                              hipStream_t stream)
{
    const float* x = (const float*)d_in[0];
    const float* L = (const float*)d_in[1];
    const float* R = (const float*)d_in[2];
    float* out = (float*)d_out;
    float* ws  = (float*)d_ws;

    const long Btot = (long)in_sizes[0] / NN;   // 4096 batch rows

    // Chunk the batch so the fp32 intermediate fits in d_ws.
    long chunk = (long)(ws_size / ((size_t)NN * sizeof(float)));
    chunk &= ~(long)(TB - 1);
    if (chunk < TB) chunk = TB;
    if (chunk > Btot) chunk = Btot;

    const dim3 blk(256);
    for (long c0 = 0; c0 < Btot; c0 += chunk) {
        long bc = Btot - c0;
        if (bc > chunk) bc = chunk;
        const dim3 grid((unsigned)(bc / TB), S / BG);
        // Stage 1: T[b][m][j] = blockdiag(L) on transposed view of x
        monarch_stage<false, false><<<grid, blk, 0, stream>>>(x + c0 * NN, L, ws);
        // Stage 2: out = blockdiag(R) on transposed view of T
        monarch_stage<true, true><<<grid, blk, 0, stream>>>(ws, R, out + c0 * NN);
    }
}